// LatentModule_8589934592319
// MI455X (gfx1250) — compile-verified
//
#include <hip/hip_runtime.h>
#include <hip/hip_bf16.h>
#include <stdint.h>

// ---------------- problem constants ----------------
#define HID   8192
#define BATCH 512

typedef __attribute__((ext_vector_type(2))) float v2f;
typedef __attribute__((ext_vector_type(4))) float f4;
typedef __attribute__((ext_vector_type(8))) float v8f;

// randomized-response probabilities (double-precision derived):
// P1_EVEN = 1.5/2.5 ; P1_ODD = 1/(1+1.5^3) ; P0 = 1/(1+1.5*exp(4/65536))
#define P1_EVEN 0.6f
#define P1_ODD  0.22857142857142857f
#define P0_ANY  0.3999853516519f

// =====================================================================
// Kernel 1: per-row Σx and Σx² via V_WMMA_F32_16X16X4_F32 with B = ones.
// grid = (4 column-blocks, 32 row-groups of 16), block = 256 (8 waves).
// Each wave owns 256 columns of a 16-row stripe: 64 iters × 2 WMMAs.
// A layout (32-bit 16x4): lanes 0-15 rows M, VGPR0=K0,VGPR1=K1;
//                         lanes 16-31 rows M, VGPR0=K2,VGPR1=K3.
// =====================================================================
__global__ void stats_wmma_kernel(const float* __restrict__ x,
                                  float* __restrict__ part) {
  const int r0     = blockIdx.y * 16;
  const int colblk = blockIdx.x;                  // 0..3 -> 2048 cols each
  const int wv     = threadIdx.x >> 5;            // wave 0..7
  const int ln     = threadIdx.x & 31;
  const int cbase  = colblk * 2048 + wv * 256;
  const int row    = r0 + (ln & 15);
  const float* base = x + (size_t)row * HID + cbase + ((ln >> 4) << 1);

  const v2f bones = {1.0f, 1.0f};
  v8f accS = {};   // Σx   (every column of D holds the row sum)
  v8f accQ = {};   // Σx²
#pragma unroll 4
  for (int kb = 0; kb < 256; kb += 4) {
    v2f a  = *(const v2f*)(base + kb);
    v2f a2 = a * a;
    accS = __builtin_amdgcn_wmma_f32_16x16x4_f32(
        false, a,  false, bones, (short)0, accS, false, false);
    accQ = __builtin_amdgcn_wmma_f32_16x16x4_f32(
        false, a2, false, bones, (short)0, accQ, false, false);
  }

  // C/D layout: lane L, VGPR v -> M = v + 8*(L>>4), N = L&15.
  // Take the N==0 column (lanes 0 and 16) and combine waves through LDS.
  __shared__ float sred[2][8][16];
  if ((ln & 15) == 0) {
    const int half = (ln >> 4) * 8;
#pragma unroll
    for (int v = 0; v < 8; ++v) {
      sred[0][wv][half + v] = accS[v];
      sred[1][wv][half + v] = accQ[v];
    }
  }
  __syncthreads();
  if (threadIdx.x < 16) {
    float s = 0.f, q = 0.f;
#pragma unroll
    for (int w = 0; w < 8; ++w) {
      s += sred[0][w][threadIdx.x];
      q += sred[1][w][threadIdx.x];
    }
    const int r = r0 + threadIdx.x;
    // deterministic per-block slots (no float atomics -> bitwise replayable)
    part[(colblk * 2 + 0) * BATCH + r] = s;
    part[(colblk * 2 + 1) * BATCH + r] = q;
  }
}

// =====================================================================
// Kernel 2: reduce 4 column-block partials -> mean, 1/sqrt(var), ddof=1
// =====================================================================
__global__ void finalize_stats_kernel(const float* __restrict__ part,
                                      float* __restrict__ stats) {
  const int r = blockIdx.x * blockDim.x + threadIdx.x;
  if (r >= BATCH) return;
  float s = 0.f, q = 0.f;
#pragma unroll
  for (int c = 0; c < 4; ++c) {
    s += part[(c * 2 + 0) * BATCH + r];
    q += part[(c * 2 + 1) * BATCH + r];
  }
  const float mean = s * (1.0f / (float)HID);
  const float var  = (q - s * s * (1.0f / (float)HID)) * (1.0f / (float)(HID - 1));
  stats[2 * r + 0] = mean;
  stats[2 * r + 1] = 1.0f / sqrtf(var);
}

// =====================================================================
// Kernel 3: streaming randomized response. Bandwidth-bound: 256 MiB of
// NT traffic (u in, out out). One thread = 8 hidden values = 64 outputs.
// All bit extraction folded into k = (uint)(16*|xn|):
//   whole_i = (k>>(4+i))&1 , frac_i = (k>>(4-i))&1  (exact: x2^n is exact)
// =====================================================================
__global__ void rr_kernel(const float* __restrict__ x,
                          const float* __restrict__ u,
                          const float* __restrict__ stats,
                          float* __restrict__ out) {
  const int row  = blockIdx.y;
  const float mean = stats[2 * row + 0];
  const float inv  = stats[2 * row + 1];
  const int h0 = (blockIdx.x * blockDim.x + threadIdx.x) * 8;

  const f4* xv = (const f4*)(x + (size_t)row * HID + h0);
  const f4* uv = (const f4*)(u + ((size_t)row * HID + h0) * 8);
  f4*       ov = (f4*)(out + ((size_t)row * HID + h0) * 8);

#pragma unroll
  for (int i = 0; i < 2; ++i) {
    const f4 xq = xv[i];                     // x may hit L2 (resident, 16 MiB)
#pragma unroll
    for (int e = 0; e < 4; ++e) {
      const float xn = (xq[e] - mean) * inv;
      const float a  = fabsf(xn);
      const uint32_t k = (uint32_t)(a * 16.0f);
      float p[8];
      p[0] = (xn < 0.0f) ? P1_EVEN : P0_ANY;   // sign bit, even position
      p[1] = (k & 16u) ? P1_ODD  : P0_ANY;     // floor(a)%2
      p[2] = (k & 32u) ? P1_EVEN : P0_ANY;     // floor(a/2)%2
      p[3] = (k & 64u) ? P1_ODD  : P0_ANY;     // floor(a/4)%2
      p[4] = (k & 8u)  ? P1_EVEN : P0_ANY;     // floor(2a)%2
      p[5] = (k & 4u)  ? P1_ODD  : P0_ANY;     // floor(4a)%2
      p[6] = (k & 2u)  ? P1_EVEN : P0_ANY;     // floor(8a)%2
      p[7] = (k & 1u)  ? P1_ODD  : P0_ANY;     // floor(16a)%2
      const int hi = i * 4 + e;
#pragma unroll
      for (int c = 0; c < 2; ++c) {
        const f4 uu = __builtin_nontemporal_load(&uv[hi * 2 + c]);
        f4 o;
        o.x = (uu.x <= p[c * 4 + 0]) ? 1.0f : 0.0f;
        o.y = (uu.y <= p[c * 4 + 1]) ? 1.0f : 0.0f;
        o.z = (uu.z <= p[c * 4 + 2]) ? 1.0f : 0.0f;
        o.w = (uu.w <= p[c * 4 + 3]) ? 1.0f : 0.0f;
        __builtin_nontemporal_store(o, &ov[hi * 2 + c]);
      }
    }
  }
}

// =====================================================================
extern "C" void kernel_launch(void* const* d_in, const int* in_sizes, int n_in,
                              void* d_out, int out_size, void* d_ws, size_t ws_size,
                              hipStream_t stream) {
  (void)in_sizes; (void)n_in; (void)out_size; (void)ws_size;
  const float* x = (const float*)d_in[0];
  const float* u = (const float*)d_in[1];
  float* out   = (float*)d_out;
  float* part  = (float*)d_ws;          // 4 colblocks * 2 * 512 = 4096 floats
  float* stats = part + 4096;           // 512 * {mean, inv_std} = 1024 floats

  // 1) per-row sums via WMMA (grid: 4 column blocks x 32 row groups)
  stats_wmma_kernel<<<dim3(4, 32), dim3(256), 0, stream>>>(x, part);
  // 2) finalize mean / inv_std
  finalize_stats_kernel<<<dim3(2), dim3(256), 0, stream>>>(part, stats);
  // 3) streaming randomized response (grid: 4 x 512, 8 h per thread)
  rr_kernel<<<dim3(4, BATCH), dim3(256), 0, stream>>>(x, u, stats, out);
}